// EdgePropertyPredictor_68839735820406
// MI455X (gfx1250) — compile-verified
//
#include <hip/hip_runtime.h>
#include <math.h>

typedef __attribute__((ext_vector_type(2))) float v2f;
typedef __attribute__((ext_vector_type(8))) float v8f;

#define E_TOTAL   250000
#define DNODE     128
#define DEMB      200
#define DSTATIC   64
#define FDIM      520          // 2*128 + 200 + 64
#define FPAD      532          // LDS row stride (floats): 532 mod 64 = 20 -> conflict-free A loads
#define NTILES    33           // ceil(520/16) N-tiles (cols 520..527 are dead padding)
#define EDGES_WG  64
#define THREADS   256          // 8 waves -> 2 waves/SIMD even at 1 WG/WGP (LDS-limited)
#define NWAVES    (THREADS / 32)
#define DOUT      8

__global__ __launch_bounds__(THREADS)
void edge_mlp_wmma_kernel(const float* __restrict__ X,
                          const int*   __restrict__ edge_index,
                          const float* __restrict__ emb,
                          const float* __restrict__ stat,
                          const float* __restrict__ W1, const float* __restrict__ b1,
                          const float* __restrict__ W2, const float* __restrict__ b2,
                          const float* __restrict__ Wout, const float* __restrict__ bout,
                          float* __restrict__ out)
{
    extern __shared__ float lds[];
    float* hA = lds;                       // 64 x 532
    float* hB = lds + EDGES_WG * FPAD;     // 64 x 532

    const int t    = threadIdx.x;
    const int lane = t & 31;
    const int wave = t >> 5;
    const int e0   = blockIdx.x * EDGES_WG;

    // ---------------- Phase 1: gather h0 = [X[src] | X[dst] | emb | static | 0-pad] ----------------
    for (int el = 0; el < EDGES_WG; ++el) {
        const int  e     = e0 + el;
        const bool valid = (e < E_TOTAL);
        const int  src   = valid ? edge_index[e] : 0;
        const int  dst   = valid ? edge_index[E_TOTAL + e] : 0;
        float* row = hA + el * FPAD;
        for (int c = t; c < FPAD; c += THREADS) {
            float v = 0.0f;
            if (valid) {
                if      (c < DNODE)                 v = X[src * DNODE + c];
                else if (c < 2 * DNODE)             v = X[dst * DNODE + (c - DNODE)];
                else if (c < 2 * DNODE + DEMB)      v = emb[e * DEMB + (c - 2 * DNODE)];
                else if (c < FDIM)                  v = stat[e * DSTATIC + (c - 2 * DNODE - DEMB)];
            }
            row[c] = v;
        }
    }
    __syncthreads();

    // ---------------- Phase 2+3: two residual layers  h = ELU(h @ W + b) + h ----------------
    const int mrow  = lane & 15;          // M (for A) / N (for B,C)
    const int khalf = (lane >> 4) << 1;   // lanes 16..31 handle K+2

    for (int L = 0; L < 2; ++L) {
        const float* W    = (L == 0) ? W1 : W2;
        const float* bias = (L == 0) ? b1 : b2;
        const float* hin  = (L == 0) ? hA : hB;
        float*       hout = (L == 0) ? hB : hA;

        for (int nt = wave; nt < NTILES; nt += NWAVES) {
            const int n   = nt * 16 + mrow;
            const int ncl = (n < FDIM) ? n : (FDIM - 1);   // clamp dead columns

            v8f acc[4];
            #pragma unroll
            for (int ms = 0; ms < 4; ++ms) acc[ms] = (v8f){0,0,0,0,0,0,0,0};

            #pragma unroll 2
            for (int k0 = 0; k0 < FDIM; k0 += 4) {
                const int kB = k0 + khalf;
                v2f bf;
                bf.x = W[(size_t)kB       * FDIM + ncl];
                bf.y = W[(size_t)(kB + 1) * FDIM + ncl];

                #pragma unroll
                for (int ms = 0; ms < 4; ++ms) {
                    const float* ap = hin + (ms * 16 + mrow) * FPAD + kB;
                    v2f af;
                    af.x = ap[0];
                    af.y = ap[1];
                    acc[ms] = __builtin_amdgcn_wmma_f32_16x16x4_f32(
                        false, af, false, bf, (short)0, acc[ms], false, false);
                }
            }

            // epilogue: bias + ELU + residual, write h_next
            const float bn = (n < FDIM) ? bias[n] : 0.0f;
            #pragma unroll
            for (int ms = 0; ms < 4; ++ms) {
                #pragma unroll
                for (int v = 0; v < 8; ++v) {
                    const int row = ms * 16 + ((lane >> 4) << 3) + v;
                    float x = acc[ms][v] + bn;
                    x = (x > 0.0f) ? x : (expf(x) - 1.0f);   // ELU
                    hout[row * FPAD + n] = x + hin[row * FPAD + n];
                }
            }
        }
        __syncthreads();
    }

    // ---------------- Phase 4: out = h @ Wout + bout  (520x8, tiny) ----------------
    for (int idx = t; idx < EDGES_WG * DOUT; idx += THREADS) {
        const int el = idx >> 3;
        const int oc = idx & 7;
        const int e  = e0 + el;
        if (e >= E_TOTAL) continue;
        const float* hrow = hA + el * FPAD;   // h2 lives in buffer A after layer 2
        float s = bout[oc];
        for (int k = 0; k < FDIM; ++k) s += hrow[k] * Wout[k * DOUT + oc];
        out[e * DOUT + oc] = s;
    }
}

extern "C" void kernel_launch(void* const* d_in, const int* in_sizes, int n_in,
                              void* d_out, int out_size, void* d_ws, size_t ws_size,
                              hipStream_t stream) {
    (void)in_sizes; (void)n_in; (void)out_size; (void)d_ws; (void)ws_size;
    const float* X    = (const float*)d_in[0];
    const int*   ei   = (const int*)  d_in[1];
    const float* emb  = (const float*)d_in[2];
    const float* stat = (const float*)d_in[3];
    const float* W1   = (const float*)d_in[4];
    const float* b1   = (const float*)d_in[5];
    const float* W2   = (const float*)d_in[6];
    const float* b2   = (const float*)d_in[7];
    const float* Wout = (const float*)d_in[8];
    const float* bout = (const float*)d_in[9];
    float* out = (float*)d_out;

    const int grid = (E_TOTAL + EDGES_WG - 1) / EDGES_WG;   // 3907
    const size_t shmem = (size_t)2 * EDGES_WG * FPAD * sizeof(float);  // 272,384 B (<= 320KB WGP LDS)
    edge_mlp_wmma_kernel<<<grid, THREADS, shmem, stream>>>(
        X, ei, emb, stat, W1, b1, W2, b2, Wout, bout, out);
}